// SubQuadraticAttention_82575041233211
// MI455X (gfx1250) — compile-verified
//
#include <hip/hip_runtime.h>

// SubQuadraticAttention for MI455X (gfx1250, wave32, WMMA + TDM).
// d_out = [ out: B*Q*HID f32 ][ attn: B*H*Q*K f32 ]
// Attention is fused (scores -> mask -> softmax -> P@V) with the 16x1024
// score tile resident in LDS, so the 268MB attn matrix is written exactly
// once (HBM roofline floor). Projections are TDM-staged WMMA GEMMs.

#define B_    4
#define Q_    1024
#define K_    1024
#define H_    16
#define HID_  1024
#define C_    64
#define TOPK_ 8
#define D_    64
#define SCALE_ 0.125f   // D^-0.5

#define PSTR_ 1044      // P row stride: 1044 % 64 == 20 -> conflict-free cols

typedef __attribute__((ext_vector_type(2))) float v2f;
typedef __attribute__((ext_vector_type(8))) float v8f;
typedef __attribute__((ext_vector_type(4))) unsigned int u32x4;
typedef __attribute__((ext_vector_type(8))) unsigned int u32x8;

// ---------------------------------------------------------------------------
// Tensor Data Mover: DMA a 2D f32 tile (tile_w x tile_h, row stride
// row_stride_elts) from global into LDS at lds_byte_off, inserting
// pad_amount DWORDs of LDS padding every 2^(pad_interval_enc+1) DWORDs.
// Per-wave scalar op; EXEC ignored; tracked by TENSORcnt. Descriptor per
// CDNA5 ISA (2D: VADDR2/3 = NULL).
// ---------------------------------------------------------------------------
__device__ __forceinline__ void tdm_load_tile_f32(
    unsigned lds_byte_off, const float* gsrc,
    unsigned tile_w, unsigned tile_h, unsigned long long row_stride_elts,
    unsigned pad_interval_enc, unsigned pad_amount_enc)
{
  const unsigned long long ga = (unsigned long long)gsrc;
  u32x4 g0;
  g0[0] = 1u;                                        // count=1, user mode
  g0[1] = lds_byte_off;                              // lds_addr
  g0[2] = (unsigned)ga;                              // global_addr[31:0]
  g0[3] = (unsigned)((ga >> 32) & 0x01ffffffu)       // global_addr[56:32]
        | 0x80000000u;                               // type=2 ("image")
  u32x8 g1;
  g1[0] = (2u << 16)                                 // data_size = 4 bytes
        | (1u << 20)                                 // pad_enable
        | (pad_interval_enc << 22)
        | (pad_amount_enc << 25);
  g1[1] = (tile_w & 0xffffu) << 16;                  // tensor_dim0 lo (=tile_w)
  g1[2] = (tile_h & 0xffffu) << 16;                  // dim0 hi=0 | tensor_dim1 lo
  g1[3] = (tile_w & 0xffffu) << 16;                  // dim1 hi=0 | tile_dim0
  g1[4] = tile_h & 0xffffu;                          // tile_dim1 | tile_dim2=0
  g1[5] = (unsigned)row_stride_elts;                 // tensor_dim0_stride lo
  g1[6] = (unsigned)((row_stride_elts >> 32) & 0xffffu);
  g1[7] = 0u;                                        // tensor_dim1_stride (2D)
  asm volatile("tensor_load_to_lds %0, %1" :: "s"(g0), "s"(g1) : "memory");
}

// ---------------------------------------------------------------------------
// Generic C = A @ W + bias GEMM. Row-major, M,N multiples of 64, Kd mult of 16.
// Block = 256 threads = 8 waves; block tile 64x64; wave tile 16x32.
// Tiles staged to LDS by the TDM (wave 0 issues, TENSORcnt-gated, then
// workgroup barrier). f32 WMMA 16x16x4 fragment layouts per ISA 7.12.2.
// ---------------------------------------------------------------------------
__global__ __launch_bounds__(256) void sqa_gemm_bias(
    const float* __restrict__ A, const float* __restrict__ W,
    const float* __restrict__ bias, float* __restrict__ Cmat,
    int M, int N, int Kd)
{
  __shared__ float As[64][20];   // 16 DW rows + 4 DW pad (TDM pad 3/3)
  __shared__ float Ws[16][68];   // 64 DW rows + 4 DW pad (TDM pad 5/3)
  const int tid  = threadIdx.x;
  const int wave = tid >> 5, lane = tid & 31;
  const int row_a = lane & 15, half = lane >> 4;
  const int m0 = blockIdx.y * 64, n0 = blockIdx.x * 64;
  const int wm = (wave & 3) * 16, wn = (wave >> 2) * 32;

  v8f acc0 = {}; v8f acc1 = {};

  for (int k0 = 0; k0 < Kd; k0 += 16) {
    if (wave == 0) {
      tdm_load_tile_f32((unsigned)(unsigned long long)&As[0][0],
                        A + (size_t)m0 * Kd + k0, 16, 64,
                        (unsigned long long)Kd, /*16 DW*/3, /*4 DW*/3);
      tdm_load_tile_f32((unsigned)(unsigned long long)&Ws[0][0],
                        W + (size_t)k0 * N + n0, 64, 16,
                        (unsigned long long)N, /*64 DW*/5, /*4 DW*/3);
      __builtin_amdgcn_s_wait_tensorcnt(0);
    }
    __syncthreads();
#pragma unroll
    for (int kk = 0; kk < 16; kk += 4) {
      v2f a, b0, b1;
      a.x  = As[wm + row_a][kk + 2 * half + 0];
      a.y  = As[wm + row_a][kk + 2 * half + 1];
      b0.x = Ws[kk + 2 * half + 0][wn + row_a];
      b0.y = Ws[kk + 2 * half + 1][wn + row_a];
      b1.x = Ws[kk + 2 * half + 0][wn + 16 + row_a];
      b1.y = Ws[kk + 2 * half + 1][wn + 16 + row_a];
      acc0 = __builtin_amdgcn_wmma_f32_16x16x4_f32(false, a, false, b0, (short)0, acc0, false, false);
      acc1 = __builtin_amdgcn_wmma_f32_16x16x4_f32(false, a, false, b1, (short)0, acc1, false, false);
    }
    __syncthreads();
  }

  const int crow = m0 + wm + half * 8;
  const int c0 = n0 + wn + row_a;
  const float bv0 = bias[c0], bv1 = bias[c0 + 16];
#pragma unroll
  for (int r = 0; r < 8; ++r) {
    Cmat[(size_t)(crow + r) * N + c0]      = acc0[r] + bv0;
    Cmat[(size_t)(crow + r) * N + c0 + 16] = acc1[r] + bv1;
  }
}

// ---------------------------------------------------------------------------
// L2-normalize centroids. One thread per (h,c) row of 64.
// ---------------------------------------------------------------------------
__global__ void sqa_centnorm(const float* __restrict__ centroids,
                             float* __restrict__ cent_n)
{
  const int t = blockIdx.x * blockDim.x + threadIdx.x;
  if (t >= H_ * C_) return;
  const float* c = centroids + (size_t)t * D_;
  float ss = 0.f;
  for (int d = 0; d < D_; ++d) ss += c[d] * c[d];
  const float inv = 1.0f / fmaxf(sqrtf(ss), 1e-12f);
  float* o = cent_n + (size_t)t * D_;
  for (int d = 0; d < D_; ++d) o[d] = c[d] * inv;
}

__global__ void sqa_zero(float* __restrict__ p, int n)
{
  const int t = blockIdx.x * blockDim.x + threadIdx.x;
  if (t < n) p[t] = 0.f;
}

// ---------------------------------------------------------------------------
// Cluster assignment: argmax_c  k . cent_n  (k-normalization drops out of
// argmax). Accumulates per-cluster counts and k-sums via atomics.
// ---------------------------------------------------------------------------
__global__ void sqa_assign(const float* __restrict__ kbuf,
                           const float* __restrict__ cent_n,
                           int* __restrict__ assign,
                           float* __restrict__ counts,
                           float* __restrict__ ksum)
{
  const int t = blockIdx.x * blockDim.x + threadIdx.x;
  if (t >= B_ * H_ * K_) return;
  const int bh = t / K_, kk = t % K_;
  const int b = bh / H_, h = bh % H_;
  const float* kv = kbuf + ((size_t)b * K_ + kk) * HID_ + h * D_;
  float kreg[D_];
  for (int d = 0; d < D_; ++d) kreg[d] = kv[d];

  int best = 0; float bestv = -__builtin_inff();
  for (int c = 0; c < C_; ++c) {
    const float* cn = cent_n + ((size_t)h * C_ + c) * D_;
    float s = 0.f;
    for (int d = 0; d < D_; ++d) s += kreg[d] * cn[d];
    if (s > bestv) { bestv = s; best = c; }
  }
  assign[t] = best;
  atomicAdd(&counts[(size_t)bh * C_ + best], 1.0f);
  float* ks = ksum + ((size_t)bh * C_ + best) * D_;
  for (int d = 0; d < D_; ++d) atomicAdd(&ks[d], kreg[d]);
}

// ccent = counts>0 ? ksum/counts : learned centroid.
__global__ void sqa_ccent(const float* __restrict__ ksum,
                          const float* __restrict__ counts,
                          const float* __restrict__ centroids,
                          float* __restrict__ ccent)
{
  const int t = blockIdx.x * blockDim.x + threadIdx.x;
  if (t >= B_ * H_ * C_ * D_) return;
  const int d = t % D_;
  const int bhc = t / D_;
  const int c = bhc % C_;
  const int h = (bhc / C_) % H_;
  const float cnt = counts[bhc];
  ccent[t] = (cnt > 0.f) ? ksum[t] / cnt
                         : centroids[((size_t)h * C_ + c) * D_ + d];
}

// ---------------------------------------------------------------------------
// Routing: per query 64 cluster scores -> top-8 (first-occurrence tie-break)
// -> uint64 cluster mask; empty-selection fallback folded in (all bits set).
// Block = 64 threads (one per cluster), grid = (Q, B*H).
// ---------------------------------------------------------------------------
__global__ __launch_bounds__(64) void sqa_route(
    const float* __restrict__ qbuf, const float* __restrict__ ccent,
    const float* __restrict__ counts, unsigned long long* __restrict__ selmask)
{
  __shared__ float sc[C_];
  const int q = blockIdx.x, bh = blockIdx.y;
  const int b = bh / H_, h = bh % H_;
  const int c = threadIdx.x;
  const float* qv = qbuf + ((size_t)b * Q_ + q) * HID_ + h * D_;
  const float* cc = ccent + ((size_t)bh * C_ + c) * D_;
  float s = 0.f;
  for (int d = 0; d < D_; ++d) s += qv[d] * cc[d];
  sc[c] = s * SCALE_;
  __syncthreads();
  if (c == 0) {
    unsigned long long m = 0;
    for (int t = 0; t < TOPK_; ++t) {
      int best = 0; float bv = -__builtin_inff();
      for (int i = 0; i < C_; ++i)
        if (!((m >> i) & 1ull) && sc[i] > bv) { bv = sc[i]; best = i; }
      m |= 1ull << best;
    }
    float csum = 0.f;
    for (int i = 0; i < C_; ++i)
      if ((m >> i) & 1ull) csum += counts[(size_t)bh * C_ + i];
    selmask[(size_t)bh * Q_ + q] = (csum > 0.f) ? m : ~0ull;
  }
}

// ---------------------------------------------------------------------------
// Fused attention: one block per (b,h, 16-query tile).
//  phase 1: masked, scaled QK^T scores -> LDS P[16][1024] (8 waves x 8 tiles,
//           16 f32-WMMA each)
//  phase 2: row softmax in LDS; normalized attn written to global ONCE
//  phase 3: out = P @ V via WMMA (A-frags from LDS, V from L2);
//           8 waves = 4 n-tiles x 2 K-halves, LDS pair-reduction.
// LDS ~72KB (CDNA5 WGP has 320KB).
// ---------------------------------------------------------------------------
__global__ __launch_bounds__(256) void sqa_attn_fused(
    const float* __restrict__ qbuf, const float* __restrict__ kbuf,
    const float* __restrict__ vbuf, const int* __restrict__ assign,
    const unsigned long long* __restrict__ selmask,
    float* __restrict__ attn, float* __restrict__ outh)
{
  __shared__ float P[16][PSTR_];
  __shared__ float red[16][17];
  __shared__ float rowmax[16];
  __shared__ float rowsum[16];
  __shared__ float Rs[4][256];

  const int tid = threadIdx.x;
  const int wave = tid >> 5, lane = tid & 31;
  const int row_a = lane & 15, half = lane >> 4;
  const int bh = blockIdx.y, b = bh >> 4, h = bh & 15;
  const int q0 = blockIdx.x * 16;

  // ---- phase 1: masked scores ----
  const float* qrow = qbuf + ((size_t)b * Q_ + q0 + row_a) * HID_ + h * D_ + 2 * half;
  const unsigned long long* smk = selmask + (size_t)bh * Q_ + q0 + half * 8;
  unsigned long long smv[8];
#pragma unroll
  for (int r = 0; r < 8; ++r) smv[r] = smk[r];

  for (int i = 0; i < 8; ++i) {
    const int kt = wave * 128 + i * 16;
    const float* krow = kbuf + ((size_t)b * K_ + kt + row_a) * HID_ + h * D_ + 2 * half;
    v8f acc = {};
#pragma unroll
    for (int d0 = 0; d0 < D_; d0 += 4) {
      v2f a, bb;
      a.x  = qrow[d0]; a.y  = qrow[d0 + 1];
      bb.x = krow[d0]; bb.y = krow[d0 + 1];
      acc = __builtin_amdgcn_wmma_f32_16x16x4_f32(false, a, false, bb, (short)0, acc, false, false);
    }
    const int kcol = kt + row_a;
    const int cl = assign[(size_t)bh * K_ + kcol];
#pragma unroll
    for (int r = 0; r < 8; ++r) {
      const bool allowed = (smv[r] >> cl) & 1ull;
      P[half * 8 + r][kcol] = allowed ? acc[r] * SCALE_ : -__builtin_inff();
    }
  }
  __syncthreads();

  // ---- phase 2: softmax (16 threads per row), single attn write ----
  {
    const int r = tid >> 4, c = tid & 15;
    float lm = -__builtin_inff();
    for (int j = 0; j < 64; ++j) lm = fmaxf(lm, P[r][c + 16 * j]);
    red[r][c] = lm; __syncthreads();
    if (c == 0) {
      float m = -__builtin_inff();
      for (int i = 0; i < 16; ++i) m = fmaxf(m, red[r][i]);
      rowmax[r] = m;
    }
    __syncthreads();
    const float m = rowmax[r];
    float ls = 0.f;
    for (int j = 0; j < 64; ++j) {
      const float e = expf(P[r][c + 16 * j] - m);
      P[r][c + 16 * j] = e;
      ls += e;
    }
    red[r][c] = ls; __syncthreads();
    if (c == 0) {
      float s = 0.f;
      for (int i = 0; i < 16; ++i) s += red[r][i];
      rowsum[r] = s;
    }
    __syncthreads();
    const float inv = 1.0f / rowsum[r];
    float* arow = attn + ((size_t)bh * Q_ + q0 + r) * K_;
    for (int j = 0; j < 64; ++j) {
      const float p = P[r][c + 16 * j] * inv;
      P[r][c + 16 * j] = p;
      arow[c + 16 * j] = p;
    }
  }
  __syncthreads();

  // ---- phase 3: out = P @ V ----
  const int n0 = (wave & 3) * 16;
  const int kh = wave >> 2;
  const float* Vb = vbuf + (size_t)b * K_ * HID_ + h * D_ + n0;
  v8f acc = {};
  for (int k = kh * 512; k < kh * 512 + 512; k += 4) {
    v2f a, bb;
    a.x  = P[row_a][k + 2 * half];
    a.y  = P[row_a][k + 2 * half + 1];
    bb.x = Vb[(size_t)(k + 2 * half) * HID_ + row_a];
    bb.y = Vb[(size_t)(k + 2 * half + 1) * HID_ + row_a];
    acc = __builtin_amdgcn_wmma_f32_16x16x4_f32(false, a, false, bb, (short)0, acc, false, false);
  }
  if (kh == 1) {
#pragma unroll
    for (int r = 0; r < 8; ++r) Rs[wave & 3][lane + 32 * r] = acc[r];
  }
  __syncthreads();
  if (kh == 0) {
    float* Ob = outh + (size_t)b * Q_ * HID_ + h * D_ + n0;
#pragma unroll
    for (int r = 0; r < 8; ++r) {
      const float v = acc[r] + Rs[wave][lane + 32 * r];
      Ob[(size_t)(q0 + half * 8 + r) * HID_ + row_a] = v;
    }
  }
}

// ---------------------------------------------------------------------------
extern "C" void kernel_launch(void* const* d_in, const int* in_sizes, int n_in,
                              void* d_out, int out_size, void* d_ws, size_t ws_size,
                              hipStream_t stream)
{
  const float* query     = (const float*)d_in[0];
  const float* key_t     = (const float*)d_in[1];
  const float* value     = (const float*)d_in[2];
  const float* centroids = (const float*)d_in[3];
  const float* Wq = (const float*)d_in[4];  const float* bq = (const float*)d_in[5];
  const float* Wk = (const float*)d_in[6];  const float* bk = (const float*)d_in[7];
  const float* Wv = (const float*)d_in[8];  const float* bv = (const float*)d_in[9];
  const float* Wo = (const float*)d_in[10]; const float* bo = (const float*)d_in[11];

  float* out  = (float*)d_out;                         // [B,Q,HID]
  float* attn = out + (size_t)B_ * Q_ * HID_;          // [B,H,Q,K]

  // Workspace carve-up (all slices 8B-aligned; total ~70.3 MB).
  float* qbuf   = (float*)d_ws;                        // [B,Q,HID]
  float* kbuf   = qbuf + (size_t)B_ * Q_ * HID_;       // [B,K,HID]
  float* vbuf   = kbuf + (size_t)B_ * K_ * HID_;       // [B,K,HID]
  float* outh   = vbuf + (size_t)B_ * K_ * HID_;       // [B,Q,HID] head-concat
  float* cent_n = outh + (size_t)B_ * Q_ * HID_;       // [H,C,D]
  float* ccent  = cent_n + (size_t)H_ * C_ * D_;       // [B,H,C,D]
  float* ksum   = ccent + (size_t)B_ * H_ * C_ * D_;   // [B,H,C,D]
  float* counts = ksum  + (size_t)B_ * H_ * C_ * D_;   // [B,H,C] (adjacent)
  int*   assign = (int*)(counts + (size_t)B_ * H_ * C_);          // [B,H,K]
  unsigned long long* selmask =
      (unsigned long long*)(assign + (size_t)B_ * H_ * K_);       // [B,H,Q]

  const dim3 blk256(256);
  const dim3 ggrid(HID_ / 64, (B_ * Q_) / 64);

  // 1) projections (WMMA GEMMs, TDM-staged)
  sqa_gemm_bias<<<ggrid, blk256, 0, stream>>>(query, Wq, bq, qbuf, B_ * Q_, HID_, HID_);
  sqa_gemm_bias<<<ggrid, blk256, 0, stream>>>(key_t, Wk, bk, kbuf, B_ * K_, HID_, HID_);
  sqa_gemm_bias<<<ggrid, blk256, 0, stream>>>(value, Wv, bv, vbuf, B_ * K_, HID_, HID_);

  // 2) clustering
  const int zn = B_ * H_ * C_ * D_ + B_ * H_ * C_;     // ksum + counts
  sqa_zero<<<dim3((zn + 255) / 256), blk256, 0, stream>>>(ksum, zn);
  sqa_centnorm<<<dim3((H_ * C_ + 255) / 256), blk256, 0, stream>>>(centroids, cent_n);
  sqa_assign<<<dim3((B_ * H_ * K_ + 255) / 256), blk256, 0, stream>>>(
      kbuf, cent_n, assign, counts, ksum);
  sqa_ccent<<<dim3((B_ * H_ * C_ * D_ + 255) / 256), blk256, 0, stream>>>(
      ksum, counts, centroids, ccent);

  // 3) top-k cluster routing -> per-query 64-bit cluster mask (w/ fallback)
  sqa_route<<<dim3(Q_, B_ * H_), dim3(64), 0, stream>>>(qbuf, ccent, counts, selmask);

  // 4) fused attention (scores+mask+softmax+P@V), attn written once
  sqa_attn_fused<<<dim3(Q_ / 16, B_ * H_), blk256, 0, stream>>>(
      qbuf, kbuf, vbuf, assign, selmask, attn, outh);

  // 5) output projection (WMMA GEMM, TDM-staged)
  sqa_gemm_bias<<<ggrid, blk256, 0, stream>>>(outh, Wo, bo, out, B_ * Q_, HID_, HID_);
}